// torchShiftAADisc_16604343566590
// MI455X (gfx1250) — compile-verified
//
#include <hip/hip_runtime.h>
#include <math.h>

// Problem constants (from reference setup_inputs): N=512, M=4096, d=8
#define NROWS 512
#define MLEN  4096
#define DDIM  8
#define TWO_PI 6.283185307179586f

typedef float v2f __attribute__((ext_vector_type(2)));
typedef float v8f __attribute__((ext_vector_type(8)));

__device__ __forceinline__ v8f wmma_f32(v2f a, v2f b, v8f c) {
  // V_WMMA_F32_16X16X4_F32: D = A(16x4) * B(4x16) + C(16x16), fp32 throughout
  return __builtin_amdgcn_wmma_f32_16x16x4_f32(
      /*neg_a=*/false, a, /*neg_b=*/false, b,
      /*c_mod=*/(short)0, c, /*reuse_a=*/false, /*reuse_b=*/false);
}

// --- CDNA5 async global<->LDS DMA (ASYNCcnt-tracked), GVS addressing -------
// CDNA5 ISA: LDS address taken from the VDST VGPR; global address = SGPR base + VGPR offset.
// LDS byte offset = low 32 bits of the flat pointer (LDS aperture mapping).
__device__ __forceinline__ void async_load_b128(unsigned lds_off, unsigned goff,
                                                const void* base) {
  asm volatile("global_load_async_to_lds_b128 %0, %1, %2 offset:0"
               :
               : "v"(lds_off), "v"(goff), "s"(base)
               : "memory");
}
__device__ __forceinline__ void async_store_b128(unsigned goff, unsigned lds_off,
                                                 void* base) {
  asm volatile("global_store_async_from_lds_b128 %0, %1, %2 offset:0"
               :
               : "v"(goff), "v"(lds_off), "s"(base)
               : "memory");
}
__device__ __forceinline__ void wait_asynccnt0() {
  asm volatile("s_wait_asynccnt 0x0" ::: "memory");
}
__device__ __forceinline__ unsigned lds_byte_off(const void* p) {
  return (unsigned)(uintptr_t)p;  // flat addr[31:0] == LDS byte offset
}

// ---------------------------------------------------------------------------
// Kernel 1: both softmaxes. C = softmax(C_tilde, axis=1) over N (8 rows of 512)
//           S = softmax(S_tilde, axis=1) over d (512 rows of 8)
// ---------------------------------------------------------------------------
__global__ void __launch_bounds__(512)
softmax_kernel(const float* __restrict__ C_tilde,
               const float* __restrict__ S_tilde,
               float* __restrict__ C, float* __restrict__ S) {
  __shared__ float red[512];
  const int t = threadIdx.x;

  // S: one row (8 elems) per thread
  {
    float v[DDIM];
    float mx = -INFINITY;
#pragma unroll
    for (int d = 0; d < DDIM; ++d) { v[d] = S_tilde[t * DDIM + d]; mx = fmaxf(mx, v[d]); }
    float sum = 0.f;
#pragma unroll
    for (int d = 0; d < DDIM; ++d) { v[d] = __expf(v[d] - mx); sum += v[d]; }
    float inv = 1.0f / sum;
#pragma unroll
    for (int d = 0; d < DDIM; ++d) S[t * DDIM + d] = v[d] * inv;
  }

  // C: block-wide reduction per d-row
  for (int d = 0; d < DDIM; ++d) {
    float v = C_tilde[d * NROWS + t];
    red[t] = v; __syncthreads();
    for (int off = 256; off > 0; off >>= 1) {
      if (t < off) red[t] = fmaxf(red[t], red[t + off]);
      __syncthreads();
    }
    float mx = red[0]; __syncthreads();
    float e = __expf(v - mx);
    red[t] = e; __syncthreads();
    for (int off = 256; off > 0; off >>= 1) {
      if (t < off) red[t] += red[t + off];
      __syncthreads();
    }
    float sum = red[0]; __syncthreads();
    C[d * NROWS + t] = e / sum;
  }
}

// ---------------------------------------------------------------------------
// Kernel 2: 4096-point FFT per row of X via four-step 64x64 decomposition,
// both 64x64 matmuls done with V_WMMA_F32_16X16X4_F32.
//   x[64*n1+n2] viewed as Xmat[n1][n2]
//   Y  = F64 @ Xmat            (Y[k1][n2], complex; Xmat real)
//   Yt = Y * e^{-2pi i k1 n2 / 4096}
//   Z  = Yt @ F64              (Z[k1][k2])
//   X_F[k1 + 64*k2] = Z[k1][k2]
// One block (4 waves, wave32) per row; each wave owns a 16-row output strip.
// Input row and output planes move via async global<->LDS DMA (ASYNCcnt).
// ---------------------------------------------------------------------------
__global__ void __launch_bounds__(128)
fft64x64_kernel(const float* __restrict__ X,
                float* __restrict__ XFre, float* __restrict__ XFim) {
  __shared__ float lx[64][65];
  __shared__ float lyre[64][65];
  __shared__ float lyim[64][65];

  const int n    = blockIdx.x;
  const int tid  = threadIdx.x;
  const int wave = tid >> 5;
  const int lane = tid & 31;
  const int l15  = lane & 15;
  const int koff = (lane >> 4) << 1;  // 0 (lanes 0-15) or 2 (lanes 16-31)
  const int row0 = wave << 4;
  const int arow = row0 + l15;        // A-matrix row held by this lane

  // async DMA the 16KB input row straight into LDS (4 floats / lane / issue)
  const float* xrow = X + (size_t)n * MLEN;
#pragma unroll
  for (int it = 0; it < 8; ++it) {
    int g = (it * 128 + tid) * 4;        // float index within row
    int r = g >> 6, c = g & 63;          // 16B chunk stays inside one row
    async_load_b128(lds_byte_off(&lx[r][c]), (unsigned)(g * 4), xrow);
  }
  wait_asynccnt0();
  __syncthreads();

  const v8f vzero = {0.f, 0.f, 0.f, 0.f, 0.f, 0.f, 0.f, 0.f};
  v8f accre[4], accim[4];
#pragma unroll
  for (int j = 0; j < 4; ++j) { accre[j] = vzero; accim[j] = vzero; }

  // ---- stage 1: Y = F64 @ Xmat; A = DFT matrix (sincos), B = real data ----
  for (int k = 0; k < 64; k += 4) {
    v2f are, aim;
#pragma unroll
    for (int t = 0; t < 2; ++t) {
      int p = (arow * (k + koff + t)) & 63;          // exact phase mod 64
      float ang = -(TWO_PI / 64.f) * (float)p;
      float sv, cv; __sincosf(ang, &sv, &cv);
      are[t] = cv; aim[t] = sv;
    }
#pragma unroll
    for (int j = 0; j < 4; ++j) {
      v2f b;
      b.x = lx[k + koff + 0][j * 16 + l15];
      b.y = lx[k + koff + 1][j * 16 + l15];
      accre[j] = wmma_f32(are, b, accre[j]);
      accim[j] = wmma_f32(aim, b, accim[j]);
    }
  }

  // ---- twiddle Yt = Y * e^{-2pi i k1 n2/4096}, stage into LDS ----
#pragma unroll
  for (int j = 0; j < 4; ++j) {
#pragma unroll
    for (int r = 0; r < 8; ++r) {
      int k1 = row0 + r + ((lane >> 4) << 3);        // C/D layout row
      int n2 = j * 16 + l15;                         // C/D layout col
      int p = (k1 * n2) & 4095;                      // exact phase mod 4096
      float ang = -(TWO_PI / 4096.f) * (float)p;
      float sv, cv; __sincosf(ang, &sv, &cv);
      float yr = accre[j][r], yi = accim[j][r];
      lyre[k1][n2] = yr * cv - yi * sv;
      lyim[k1][n2] = yr * sv + yi * cv;
    }
  }
  __syncthreads();

  // ---- stage 2: Z = Yt @ F64; A = complex data (LDS), B = DFT (sincos) ----
#pragma unroll
  for (int j = 0; j < 4; ++j) { accre[j] = vzero; accim[j] = vzero; }

  for (int k = 0; k < 64; k += 4) {
    v2f are, aim;
    are.x = lyre[arow][k + koff + 0];
    are.y = lyre[arow][k + koff + 1];
    aim.x = lyim[arow][k + koff + 0];
    aim.y = lyim[arow][k + koff + 1];
#pragma unroll
    for (int j = 0; j < 4; ++j) {
      v2f bre, bim, bimn;
#pragma unroll
      for (int t = 0; t < 2; ++t) {
        int col = j * 16 + l15;
        int p = ((k + koff + t) * col) & 63;
        float ang = -(TWO_PI / 64.f) * (float)p;
        float sv, cv; __sincosf(ang, &sv, &cv);
        bre[t] = cv; bim[t] = sv; bimn[t] = -sv;
      }
      // complex matmul = 4 real WMMAs (imag B pre-negated for the Re part)
      accre[j] = wmma_f32(are, bre,  accre[j]);
      accre[j] = wmma_f32(aim, bimn, accre[j]);
      accim[j] = wmma_f32(are, bim,  accim[j]);
      accim[j] = wmma_f32(aim, bre,  accim[j]);
    }
  }
  __syncthreads();   // all waves done reading ly strips before restaging

  // ---- stage Z transposed into LDS: ly[k2][k1] = Z[k1][k2] ----
  // (out flat index = 64*k2 + k1, so rows of ly[k2][*] are contiguous chunks)
#pragma unroll
  for (int j = 0; j < 4; ++j) {
#pragma unroll
    for (int r = 0; r < 8; ++r) {
      int k1 = row0 + r + ((lane >> 4) << 3);
      int k2 = j * 16 + l15;
      lyre[k2][k1] = accre[j][r];
      lyim[k2][k1] = accim[j][r];
    }
  }
  __syncthreads();

  // ---- coalesced async DMA of both planes: LDS -> global ----
  float* xfre_row = XFre + (size_t)n * MLEN;
  float* xfim_row = XFim + (size_t)n * MLEN;
#pragma unroll
  for (int it = 0; it < 8; ++it) {
    int g = (it * 128 + tid) * 4;        // out flat index = 64*k2 + k1
    int r = g >> 6, c = g & 63;
    async_store_b128((unsigned)(g * 4), lds_byte_off(&lyre[r][c]), xfre_row);
    async_store_b128((unsigned)(g * 4), lds_byte_off(&lyim[r][c]), xfim_row);
  }
  wait_asynccnt0();   // drain before wave exit (S_ENDPGM also waits idle)
}

// ---------------------------------------------------------------------------
// Kernel 3: A_F[d,m] = sum_N C[d,N] * X_F[N,m] * e^{+2pi i tau[N,d] f_m}
// One thread per m; C and tau cached in LDS (32 KB); X_F streamed from L2.
// ---------------------------------------------------------------------------
__global__ void __launch_bounds__(256)
af_kernel(const float* __restrict__ XFre, const float* __restrict__ XFim,
          const float* __restrict__ C, const float* __restrict__ tau,
          float* __restrict__ AFre, float* __restrict__ AFim) {
  __shared__ float sC[DDIM][NROWS];
  __shared__ float sT[NROWS][DDIM];
  const int tid = threadIdx.x;
  for (int i = tid; i < DDIM * NROWS; i += 256) sC[i >> 9][i & 511] = C[i];
  for (int i = tid; i < NROWS * DDIM; i += 256) sT[i >> 3][i & 7] = tau[i];
  __syncthreads();

  const int m = blockIdx.x * 256 + tid;
  const float fm = (float)m * (1.0f / (float)MLEN);

  float are[DDIM], aim[DDIM];
#pragma unroll
  for (int d = 0; d < DDIM; ++d) { are[d] = 0.f; aim[d] = 0.f; }

  for (int nn = 0; nn < NROWS; ++nn) {
    float xr = XFre[(size_t)nn * MLEN + m];
    float xi = XFim[(size_t)nn * MLEN + m];
#pragma unroll
    for (int d = 0; d < DDIM; ++d) {
      float ang = TWO_PI * sT[nn][d] * fm;   // conj(omega) = e^{+i 2pi tau f}
      float sv, cv; __sincosf(ang, &sv, &cv);
      float w = sC[d][nn];
      are[d] += w * (xr * cv - xi * sv);
      aim[d] += w * (xr * sv + xi * cv);
    }
  }
#pragma unroll
  for (int d = 0; d < DDIM; ++d) {
    AFre[d * MLEN + m] = are[d];
    AFim[d * MLEN + m] = aim[d];
  }
}

// ---------------------------------------------------------------------------
// Kernel 4: x[n,m] = sum_d S[n,d] * e^{-2pi i tau[n,d] f_m} * A_F[d,m]
// One block per row n; interleaved complex64 output.
// ---------------------------------------------------------------------------
__global__ void __launch_bounds__(256)
out_kernel(const float* __restrict__ AFre, const float* __restrict__ AFim,
           const float* __restrict__ S, const float* __restrict__ tau,
           float* __restrict__ out) {
  const int n = blockIdx.x;
  float sd[DDIM], td[DDIM];
#pragma unroll
  for (int d = 0; d < DDIM; ++d) {
    sd[d] = S[n * DDIM + d];
    td[d] = tau[n * DDIM + d];
  }
  for (int m = threadIdx.x; m < MLEN; m += 256) {
    const float fm = (float)m * (1.0f / (float)MLEN);
    float xr = 0.f, xi = 0.f;
#pragma unroll
    for (int d = 0; d < DDIM; ++d) {
      float ang = -TWO_PI * td[d] * fm;      // omega
      float sv, cv; __sincosf(ang, &sv, &cv);
      float ar = AFre[d * MLEN + m], ai = AFim[d * MLEN + m];
      float wr = sd[d] * cv, wi = sd[d] * sv;
      xr += wr * ar - wi * ai;
      xi += wr * ai + wi * ar;
    }
    size_t o = ((size_t)n * MLEN + m) * 2;   // interleaved complex64
    out[o + 0] = xr;
    out[o + 1] = xi;
  }
}

// ---------------------------------------------------------------------------
extern "C" void kernel_launch(void* const* d_in, const int* in_sizes, int n_in,
                              void* d_out, int out_size, void* d_ws, size_t ws_size,
                              hipStream_t stream) {
  const float* X       = (const float*)d_in[0];   // [512,4096]
  const float* C_tilde = (const float*)d_in[1];   // [8,512]
  const float* S_tilde = (const float*)d_in[2];   // [512,8]
  const float* tau     = (const float*)d_in[3];   // [512,8]

  float* ws   = (float*)d_ws;
  float* XFre = ws;                                   // 512*4096
  float* XFim = XFre + (size_t)NROWS * MLEN;          // 512*4096
  float* AFre = XFim + (size_t)NROWS * MLEN;          // 8*4096
  float* AFim = AFre + (size_t)DDIM * MLEN;           // 8*4096
  float* Cw   = AFim + (size_t)DDIM * MLEN;           // 8*512
  float* Sw   = Cw   + (size_t)DDIM * NROWS;          // 512*8
  float* out  = (float*)d_out;                        // complex64 -> 2*N*M floats

  softmax_kernel<<<1, 512, 0, stream>>>(C_tilde, S_tilde, Cw, Sw);
  fft64x64_kernel<<<NROWS, 128, 0, stream>>>(X, XFre, XFim);
  af_kernel<<<MLEN / 256, 256, 0, stream>>>(XFre, XFim, Cw, tau, AFre, AFim);
  out_kernel<<<NROWS, 256, 0, stream>>>(AFre, AFim, Sw, tau, out);
}